// CTCLayer_68856915690047
// MI455X (gfx1250) — compile-verified
//
#include <hip/hip_runtime.h>

// CTC batch cost, Keras convention (blank = C-1), full-length inputs.
// B=256, T=512, C=128, U=48 -> S=97. One wave32 per batch element.
// alpha recursion in log2 domain; async global->LDS pipeline (DEPTH=8).

#define NEGV  (-1.0e30f)
#define EPSV  (1e-7f)

static constexpr int Tn    = 512;
static constexpr int Cn    = 128;   // classes, blank = 127
static constexpr int Un    = 48;    // label length
static constexpr int DEPTH = 8;     // async pipeline depth (rows in flight)

typedef int v4i __attribute__((vector_size(16)));
typedef __attribute__((address_space(1))) v4i* gv4i_p;   // global int4*
typedef __attribute__((address_space(3))) v4i* lv4i_p;   // LDS int4*
typedef __attribute__((address_space(1))) void* gvoid_p;
typedef __attribute__((address_space(3))) void* lvoid_p;

__device__ __forceinline__ float fexp2(float x) {
#if __has_builtin(__builtin_amdgcn_exp2f)
  return __builtin_amdgcn_exp2f(x);      // v_exp_f32
#else
  return exp2f(x);
#endif
}

__device__ __forceinline__ float flog2(float x) {
#if __has_builtin(__builtin_amdgcn_logf)
  return __builtin_amdgcn_logf(x);       // v_log_f32
#else
  return log2f(x);
#endif
}

// log2-domain logsumexp of three terms
__device__ __forceinline__ float lse3_2(float a, float b, float c) {
  float m = fmaxf(a, fmaxf(b, c));
  float s = fexp2(a - m) + fexp2(b - m) + fexp2(c - m);
  return m + flog2(s);
}

// Copy one 512B row (128 floats) global -> LDS, asynchronously (ASYNCcnt).
// Each of the 32 lanes moves 16 bytes via b128.
__device__ __forceinline__ void async_row(const float* g, float* l) {
#if __has_builtin(__builtin_amdgcn_global_load_async_to_lds_b128)
  __builtin_amdgcn_global_load_async_to_lds_b128(
      (gv4i_p)(gvoid_p)(void*)g,
      (lv4i_p)(lvoid_p)(void*)l, 0, 0);
#else
  unsigned loff = (unsigned)(unsigned long long)l;  // low 32b = LDS offset
  asm volatile("global_load_async_to_lds_b128 %0, %1, off"
               :: "v"(loff), "v"(g) : "memory");
#endif
}

__device__ __forceinline__ void wait_async_le7() {
  asm volatile("s_wait_asynccnt 7" ::: "memory");
}

__global__ void __launch_bounds__(32)
ctc_fwd_kernel(const int* __restrict__ y_true,
               const float* __restrict__ y_pred,
               float* __restrict__ out) {
  const int b    = blockIdx.x;
  const int lane = threadIdx.x;

  __shared__ float buf[DEPTH * Cn];   // 4 KB ring of prob rows

  // ---- extended labels & skip flags for this lane ----------------------
  // s = lane + 32*g (g=0..2), parity of s == parity of lane.
  // odd s=2u+1 -> label u; even s -> blank (127). s=96 (lane0,g=3) -> blank.
  const int* lab = y_true + b * Un;
  int  e0, e1, e2;
  bool sk0 = false, sk1 = false, sk2 = false;
  if (lane & 1) {
    int u  = (lane - 1) >> 1;               // 0..15
    int l0 = lab[u], l1 = lab[u + 16], l2 = lab[u + 32];
    e0 = l0; e1 = l1; e2 = l2;
    sk0 = (u == 0) || (l0 != lab[u - 1]);
    sk1 = (l1 != lab[u + 15]);
    sk2 = (l2 != lab[u + 31]);
  } else {
    e0 = e1 = e2 = Cn - 1;                  // blank
  }

  const float* base = y_pred + (long long)b * Tn * Cn;
  const int goff = lane * 4;                // 16B per lane

  // ---- prologue: fill the async pipeline ------------------------------
  for (int d = 0; d < DEPTH; ++d)
    async_row(base + d * Cn + goff, &buf[d * Cn + goff]);

  float a0, a1, a2, a3;

  // ---- t = 0 init: only s=0 (blank) and s=1 (first label) -------------
  {
    wait_async_le7();                       // row 0 landed
    const float* row = &buf[0];
    float lp0 = flog2(row[e0] + EPSV);
    a0 = (lane < 2) ? lp0 : NEGV;
    a1 = NEGV; a2 = NEGV; a3 = NEGV;
    asm volatile("" : "+v"(lp0));           // DS reads done before reuse
    async_row(base + DEPTH * Cn + goff, &buf[0 * Cn + goff]);
  }

  // ---- main DP loop ----------------------------------------------------
  for (int t = 1; t < Tn; ++t) {
    wait_async_le7();                       // row t landed (in-order retire)
    const float* row = &buf[(t & (DEPTH - 1)) * Cn];
    float lp0 = flog2(row[e0] + EPSV);
    float lp1 = flog2(row[e1] + EPSV);
    float lp2 = flog2(row[e2] + EPSV);
    float lp3 = flog2(row[Cn - 1] + EPSV);  // s=96 blank
    asm volatile("" : "+v"(lp0), "+v"(lp1), "+v"(lp2), "+v"(lp3));
    if (t + DEPTH < Tn)                     // reuse this slot for row t+DEPTH
      async_row(base + (t + DEPTH) * Cn + goff,
                &buf[(t & (DEPTH - 1)) * Cn + goff]);

    // neighbor alphas across the 32-lane striping
    float ab0 = __shfl(a0, 31, 32), ab1 = __shfl(a1, 31, 32), ab2 = __shfl(a2, 31, 32);
    float ac0 = __shfl(a0, 30, 32), ac1 = __shfl(a1, 30, 32), ac2 = __shfl(a2, 30, 32);
    float p10 = __shfl_up(a0, 1, 32), p11 = __shfl_up(a1, 1, 32), p12 = __shfl_up(a2, 1, 32);
    float p20 = __shfl_up(a0, 2, 32), p21 = __shfl_up(a1, 2, 32), p22 = __shfl_up(a2, 2, 32);
    if (lane == 0) { p10 = NEGV; p11 = ab0; p12 = ab1;
                     p20 = NEGV; p21 = ac0; p22 = ac1; }
    if (lane == 1) { p20 = NEGV; p21 = ab0; p22 = ab1; }
    p20 = sk0 ? p20 : NEGV;
    p21 = sk1 ? p21 : NEGV;
    p22 = sk2 ? p22 : NEGV;

    float n0 = lse3_2(a0, p10, p20) + lp0;
    float n1 = lse3_2(a1, p11, p21) + lp1;
    float n2 = lse3_2(a2, p12, p22) + lp2;
    float n3 = lse3_2(a3, ab2, NEGV) + lp3; // s=96: prev1=alpha[95], no skip
    a0 = n0; a1 = n1; a2 = n2;
    a3 = (lane == 0) ? n3 : NEGV;
  }

  // ---- finalize: -lse(alpha[95], alpha[96]) * ln2 ---------------------
  float v95 = __shfl(a2, 31, 32);
  if (lane == 0) {
    float m  = fmaxf(v95, a3);
    float ll = (m + flog2(fexp2(v95 - m) + fexp2(a3 - m)))
             * 0.69314718055994530942f;     // log2 -> ln
    out[b] = -ll;
  }
}

extern "C" void kernel_launch(void* const* d_in, const int* in_sizes, int n_in,
                              void* d_out, int out_size, void* d_ws, size_t ws_size,
                              hipStream_t stream) {
  const int*   y_true = (const int*)d_in[0];    // [B, U]
  const float* y_pred = (const float*)d_in[1];  // [B, T, C]
  float*       out    = (float*)d_out;          // [B, 1]
  const int B = out_size;                       // 256
  ctc_fwd_kernel<<<B, 32, 0, stream>>>(y_true, y_pred, out);
}